// GNNModel_14783277433090
// MI455X (gfx1250) — compile-verified
//
#include <hip/hip_runtime.h>
#include <hip/hip_bf16.h>
#include <math.h>

typedef __attribute__((ext_vector_type(2))) float v2f;
typedef __attribute__((ext_vector_type(8))) float v8f;

#define NN      100000
#define NE      1600000
#define HH      64
#define TT      384
#define NTILES  6250       // NN / 16 exactly

__device__ __forceinline__ v2f mk2(float a, float b) {
    v2f r; r.x = a; r.y = b; return r;
}

__device__ __forceinline__ v8f wmma4(v2f a, v2f b, v8f c) {
    // D = A(16x4,f32) * B(4x16,f32) + C(16x16,f32)
    return __builtin_amdgcn_wmma_f32_16x16x4_f32(
        /*neg_a=*/false, a, /*neg_b=*/false, b,
        /*c_mod=*/(short)0, c, /*reuse_a=*/false, /*reuse_b=*/false);
}

// ---------------------------------------------------------------------------
// Degree counting + inversion (DGL mean: zero in-degree -> 0)
// ---------------------------------------------------------------------------
__global__ void degree_kernel(const int* __restrict__ src, const int* __restrict__ dst,
                              float* __restrict__ degF, float* __restrict__ degR) {
    int e = blockIdx.x * 256 + threadIdx.x;
    if (e < NE) {
        atomicAdd(&degF[dst[e]], 1.0f);
        atomicAdd(&degR[src[e]], 1.0f);
    }
}

__global__ void invert_kernel(float* __restrict__ degF, float* __restrict__ degR) {
    int i = blockIdx.x * 256 + threadIdx.x;
    if (i < NN) {
        float a = degF[i];
        float b = degR[i];
        degF[i] = (a > 0.0f) ? (1.0f / a) : 0.0f;
        degR[i] = (b > 0.0f) ? (1.0f / b) : 0.0f;
    }
}

// ---------------------------------------------------------------------------
// Encoder: x = text_emb[100000,384] @ W_enc[384,64] + b_enc
// 1 wave -> one 16-row node tile, 4 WMMA accumulators covering all 64 cols.
// ---------------------------------------------------------------------------
__global__ __launch_bounds__(256) void enc_kernel(const float* __restrict__ emb,
                                                  const float* __restrict__ W,
                                                  const float* __restrict__ bias,
                                                  float* __restrict__ x) {
    const int wave = threadIdx.x >> 5;
    const int lane = threadIdx.x & 31;
    const int nt   = blockIdx.x * 8 + wave;
    if (nt >= NTILES) return;
    const int ln = lane & 15;        // M for A, N for B/C/D
    const int hi = lane >> 4;        // lane half
    const int kh = hi << 1;          // K offset within 4-wide K step

    const float* arow = emb + (size_t)(nt * 16 + ln) * TT + kh;
    v8f acc0 = {}, acc1 = {}, acc2 = {}, acc3 = {};

    #pragma unroll 4
    for (int k0 = 0; k0 < TT; k0 += 4) {
        v2f a = *reinterpret_cast<const v2f*>(arow + k0);
        const float* bp = W + (size_t)(k0 + kh) * HH + ln;
        v2f b0 = mk2(bp[0],  bp[HH]);
        v2f b1 = mk2(bp[16], bp[HH + 16]);
        v2f b2 = mk2(bp[32], bp[HH + 32]);
        v2f b3 = mk2(bp[48], bp[HH + 48]);
        acc0 = wmma4(a, b0, acc0);
        acc1 = wmma4(a, b1, acc1);
        acc2 = wmma4(a, b2, acc2);
        acc3 = wmma4(a, b3, acc3);
    }

    float bs0 = bias[ln], bs1 = bias[16 + ln], bs2 = bias[32 + ln], bs3 = bias[48 + ln];
    #pragma unroll
    for (int v = 0; v < 8; ++v) {
        size_t r = (size_t)(nt * 16 + v + 8 * hi) * HH + ln;
        x[r]      = acc0[v] + bs0;
        x[r + 16] = acc1[v] + bs1;
        x[r + 32] = acc2[v] + bs2;
        x[r + 48] = acc3[v] + bs3;
    }
}

// ---------------------------------------------------------------------------
// Edge scatter: both directions in one pass.
// thread (e, c): neighF[dst[e],c] += w*x[src[e],c]; neighR[src[e],c] += w*x[dst[e],c]
// Lanes of a wave cover 32 consecutive channels of one edge -> coalesced.
// ---------------------------------------------------------------------------
__global__ __launch_bounds__(256) void scatter_kernel(const float* __restrict__ x,
                                                      const float* __restrict__ w,
                                                      const int* __restrict__ src,
                                                      const int* __restrict__ dst,
                                                      float* __restrict__ nF,
                                                      float* __restrict__ nR) {
    size_t gid = (size_t)blockIdx.x * 256 + threadIdx.x;
    int e = (int)(gid >> 6);
    int c = (int)(gid & 63);
    if (e >= NE) return;
    int s = src[e];
    int d = dst[e];
    float we = w[e];
    float xs = x[(size_t)s * HH + c];
    float xd = x[(size_t)d * HH + c];
    atomicAdd(&nF[(size_t)d * HH + c], we * xs);
    atomicAdd(&nR[(size_t)s * HH + c], we * xd);
}

// ---------------------------------------------------------------------------
// Fused layer update (in-place, node-pointwise):
//   x += relu(x@WsF + (nF*invF)@WnF + bF) + relu(x@WsR + (nR*invR)@WnR + bR)
// 1 wave per 16-node tile: 8 accumulators, 256 WMMAs.
// ---------------------------------------------------------------------------
__global__ __launch_bounds__(256) void layer_kernel(
    float* __restrict__ x,
    const float* __restrict__ nF, const float* __restrict__ nR,
    const float* __restrict__ invF, const float* __restrict__ invR,
    const float* __restrict__ WsF, const float* __restrict__ WnF, const float* __restrict__ bF,
    const float* __restrict__ WsR, const float* __restrict__ WnR, const float* __restrict__ bR) {
    const int wave = threadIdx.x >> 5;
    const int lane = threadIdx.x & 31;
    const int nt   = blockIdx.x * 8 + wave;
    if (nt >= NTILES) return;
    const int ln = lane & 15;
    const int hi = lane >> 4;
    const int kh = hi << 1;
    const int rowA = nt * 16 + ln;

    const float sF = invF[rowA];
    const float sR = invR[rowA];
    const float* xr = x  + (size_t)rowA * HH + kh;
    const float* fr = nF + (size_t)rowA * HH + kh;
    const float* rr = nR + (size_t)rowA * HH + kh;

    v8f aF0 = {}, aF1 = {}, aF2 = {}, aF3 = {};
    v8f aR0 = {}, aR1 = {}, aR2 = {}, aR3 = {};

    #pragma unroll 4
    for (int k0 = 0; k0 < HH; k0 += 4) {
        v2f aS = *reinterpret_cast<const v2f*>(xr + k0);
        v2f aN = *reinterpret_cast<const v2f*>(fr + k0);
        v2f aV = *reinterpret_cast<const v2f*>(rr + k0);
        aN *= sF;   // mean-scale forward neighbor rows (per-row scalar)
        aV *= sR;   // mean-scale reverse neighbor rows

        const float* pSF = WsF + (size_t)(k0 + kh) * HH + ln;
        const float* pNF = WnF + (size_t)(k0 + kh) * HH + ln;
        const float* pSR = WsR + (size_t)(k0 + kh) * HH + ln;
        const float* pNR = WnR + (size_t)(k0 + kh) * HH + ln;

        // column tile 0
        aF0 = wmma4(aS, mk2(pSF[0], pSF[HH]), aF0);
        aF0 = wmma4(aN, mk2(pNF[0], pNF[HH]), aF0);
        aR0 = wmma4(aS, mk2(pSR[0], pSR[HH]), aR0);
        aR0 = wmma4(aV, mk2(pNR[0], pNR[HH]), aR0);
        // column tile 1
        aF1 = wmma4(aS, mk2(pSF[16], pSF[HH + 16]), aF1);
        aF1 = wmma4(aN, mk2(pNF[16], pNF[HH + 16]), aF1);
        aR1 = wmma4(aS, mk2(pSR[16], pSR[HH + 16]), aR1);
        aR1 = wmma4(aV, mk2(pNR[16], pNR[HH + 16]), aR1);
        // column tile 2
        aF2 = wmma4(aS, mk2(pSF[32], pSF[HH + 32]), aF2);
        aF2 = wmma4(aN, mk2(pNF[32], pNF[HH + 32]), aF2);
        aR2 = wmma4(aS, mk2(pSR[32], pSR[HH + 32]), aR2);
        aR2 = wmma4(aV, mk2(pNR[32], pNR[HH + 32]), aR2);
        // column tile 3
        aF3 = wmma4(aS, mk2(pSF[48], pSF[HH + 48]), aF3);
        aF3 = wmma4(aN, mk2(pNF[48], pNF[HH + 48]), aF3);
        aR3 = wmma4(aS, mk2(pSR[48], pSR[HH + 48]), aR3);
        aR3 = wmma4(aV, mk2(pNR[48], pNR[HH + 48]), aR3);
    }

    float bf0 = bF[ln], bf1 = bF[16 + ln], bf2 = bF[32 + ln], bf3 = bF[48 + ln];
    float br0 = bR[ln], br1 = bR[16 + ln], br2 = bR[32 + ln], br3 = bR[48 + ln];

    #pragma unroll
    for (int v = 0; v < 8; ++v) {
        size_t r = (size_t)(nt * 16 + v + 8 * hi) * HH + ln;
        x[r]      += fmaxf(aF0[v] + bf0, 0.0f) + fmaxf(aR0[v] + br0, 0.0f);
        x[r + 16] += fmaxf(aF1[v] + bf1, 0.0f) + fmaxf(aR1[v] + br1, 0.0f);
        x[r + 32] += fmaxf(aF2[v] + bf2, 0.0f) + fmaxf(aR2[v] + br2, 0.0f);
        x[r + 48] += fmaxf(aF3[v] + bf3, 0.0f) + fmaxf(aR3[v] + br3, 0.0f);
    }
}

// ---------------------------------------------------------------------------
// Batch gather + L2 normalize: one 64-thread block per output row.
// ---------------------------------------------------------------------------
__global__ __launch_bounds__(64) void norm_kernel(const float* __restrict__ x,
                                                  const int* __restrict__ ids,
                                                  float* __restrict__ out) {
    __shared__ float s[64];
    int b = blockIdx.x;
    int c = threadIdx.x;
    int node = ids[b];
    float v = x[(size_t)node * HH + c];
    s[c] = v * v;
    __syncthreads();
    #pragma unroll
    for (int off = 32; off > 0; off >>= 1) {
        if (c < off) s[c] += s[c + off];
        __syncthreads();
    }
    float nrm = sqrtf(s[0]);
    out[(size_t)b * HH + c] = v / nrm;
}

// ---------------------------------------------------------------------------
// Host orchestration (graph-capture safe: only async ops on `stream`).
// Workspace layout (floats): x[NN*64] | neighF[NN*64] | neighR[NN*64] | invF[NN] | invR[NN]
// = 3*6.4M + 0.2M floats ~= 77.6 MB
// ---------------------------------------------------------------------------
extern "C" void kernel_launch(void* const* d_in, const int* in_sizes, int n_in,
                              void* d_out, int out_size, void* d_ws, size_t ws_size,
                              hipStream_t stream) {
    const float* text_emb = (const float*)d_in[0];
    const float* weights  = (const float*)d_in[1];
    const float* W_enc    = (const float*)d_in[2];
    const float* b_enc    = (const float*)d_in[3];
    const float* Wself_f  = (const float*)d_in[4];
    const float* Wneigh_f = (const float*)d_in[5];
    const float* b_f      = (const float*)d_in[6];
    const float* Wself_r  = (const float*)d_in[7];
    const float* Wneigh_r = (const float*)d_in[8];
    const float* b_r      = (const float*)d_in[9];
    const int*   src      = (const int*)d_in[10];
    const int*   dst      = (const int*)d_in[11];
    const int*   ids      = (const int*)d_in[12];
    float* out = (float*)d_out;

    const size_t NH = (size_t)NN * HH;
    float* x    = (float*)d_ws;
    float* nF   = x + NH;
    float* nR   = x + 2 * NH;
    float* invF = x + 3 * NH;
    float* invR = invF + NN;

    const int batch = in_sizes[12];

    // degrees
    hipMemsetAsync(invF, 0, (size_t)2 * NN * sizeof(float), stream);
    degree_kernel<<<(NE + 255) / 256, 256, 0, stream>>>(src, dst, invF, invR);
    invert_kernel<<<(NN + 255) / 256, 256, 0, stream>>>(invF, invR);

    // encoder
    enc_kernel<<<(NTILES + 7) / 8, 256, 0, stream>>>(text_emb, W_enc, b_enc, x);

    // layers
    for (int l = 0; l < 2; ++l) {
        hipMemsetAsync(nF, 0, (size_t)2 * NH * sizeof(float), stream);
        scatter_kernel<<<(int)(((size_t)NE * HH + 255) / 256), 256, 0, stream>>>(
            x, weights, src, dst, nF, nR);
        layer_kernel<<<(NTILES + 7) / 8, 256, 0, stream>>>(
            x, nF, nR, invF, invR,
            Wself_f + (size_t)l * HH * HH, Wneigh_f + (size_t)l * HH * HH, b_f + (size_t)l * HH,
            Wself_r + (size_t)l * HH * HH, Wneigh_r + (size_t)l * HH * HH, b_r + (size_t)l * HH);
    }

    // gather + normalize
    norm_kernel<<<batch, 64, 0, stream>>>(x, ids, out);
}